// CustomGradient_36524401885324
// MI455X (gfx1250) — compile-verified
//
#include <hip/hip_runtime.h>
#include <cmath>

typedef __attribute__((ext_vector_type(2))) float v2f;
typedef __attribute__((ext_vector_type(8))) float v8f;
typedef __attribute__((ext_vector_type(4))) int   v4i;

constexpr int kN = 128;        // image size == detector count
constexpr int kB = 8;          // batch
constexpr int kC = 128;        // channels (diagonal gather: detector row == channel)
constexpr int kA = 180;        // angles
constexpr int kThreads = 384;  // 12 wave32 -> 12 angle tiles of 16

// Zero-padded LDS tile: rays stay within |coord-63.5| <= 63.5*sqrt(2) ~ 89.8,
// so indices fall in [-27, 154]; pad 32 each side -> all taps in-bounds, and
// out-of-image taps read 0 (identical to reference's weight*valid masking).
constexpr int kPad = 32;
constexpr int kStride = 192;   // padded row length in floats (192 = 128 + 2*32)

#if defined(__has_builtin)
#if __has_builtin(__builtin_amdgcn_global_load_async_to_lds_b128)
#define USE_ASYNC_LDS 1
#endif
#endif

// Branch-free bilinear sample from the padded LDS tile.
__device__ __forceinline__ float bsample(const float* __restrict__ L,
                                         float uc, float vc,
                                         float cs, float sn, float t) {
  float u = fmaf(-t, sn, uc);            // row coord
  float v = fmaf( t, cs, vc);            // col coord
  float fu = floorf(u), fv = floorf(v);
  float wu = u - fu, wv = v - fv;
  int iu = (int)fu + kPad;
  int iv = (int)fv + kPad;
  const float* row0 = L + iu * kStride + iv;
  const float* row1 = row0 + kStride;
  float p00 = row0[0], p01 = row0[1];
  float p10 = row1[0], p11 = row1[1];
  float b0 = 1.f - wv;
  float top = fmaf(p01, wv, p00 * b0);
  float bot = fmaf(p11, wv, p10 * b0);
  return fmaf(bot, wu, top * (1.f - wu));
}

__global__ __launch_bounds__(kThreads)
void radon_diag_kernel(const float* __restrict__ X,
                       const float* __restrict__ angles,
                       float* __restrict__ out) {
  __shared__ float img[kStride * kStride];     // 147 KB of the 320 KB WGP LDS

  const int tid = threadIdx.x;
  const int c = blockIdx.x;                    // channel == detector row
  const int b = blockIdx.y;                    // batch

  float4* s4 = (float4*)img;
  const float4* __restrict__ g4 =
      (const float4*)(X + ((size_t)(b * kC + c) << 14));

  // --- Zero the border (disjoint from interior fill; 5120 float4 cells) ---
  {
    const float4 z = {0.f, 0.f, 0.f, 0.f};
    for (int i = tid; i < 5120; i += kThreads) {
      int off4;
      if (i < 1536) {                          // rows 0..31 (full)
        off4 = i;
      } else if (i < 3072) {                   // rows 160..191 (full)
        off4 = (160 * kStride / 4) + (i - 1536);
      } else {                                 // rows 32..159, cols 0..31 & 160..191
        int k = i - 3072;                      // 0..2047, 16 float4 per row
        int r = 32 + (k >> 4);
        int cc = k & 15;
        int col4 = (cc < 8) ? cc : (cc - 8 + 40);
        off4 = r * (kStride / 4) + col4;
      }
      s4[off4] = z;
    }
  }

  // --- Stage the 128x128 interior (row r -> padded row r+32, col +32) ---
#ifdef USE_ASYNC_LDS
  for (int i = tid; i < 4096; i += kThreads) {
    int r = i >> 5, c4 = i & 31;
    int dst = (r + kPad) * (kStride / 4) + (kPad / 4) + c4;
    __builtin_amdgcn_global_load_async_to_lds_b128(
        (v4i*)(g4 + i), (v4i*)(s4 + dst), 0, 0);
  }
  asm volatile("s_wait_asynccnt 0x0" ::: "memory");
#else
  for (int i = tid; i < 4096; i += kThreads) {
    int r = i >> 5, c4 = i & 31;
    s4[(r + kPad) * (kStride / 4) + (kPad / 4) + c4] = g4[i];
  }
#endif
  __syncthreads();

  // --- Per-wave: one tile of 16 angles, WMMA reduces K=4 ray samples ---
  const int lane = tid & 31;
  const int wave = tid >> 5;                   // 0..11 -> angle tile
  const int base = wave * 16;
  const int m    = lane & 15;                  // A-matrix row (angle in tile)
  const int hh   = lane >> 4;                  // lanes 0-15: K=0,1; lanes 16-31: K=2,3

  const int aidx = min(base + m, kA - 1);      // clamp padded tail angles
  float sn, cs;
  sincosf(angles[aidx], &sn, &cs);
  const float dd = (float)c - 63.5f;           // detector coordinate d = c - c0
  const float uc = fmaf(dd, cs, 63.5f);
  const float vc = fmaf(dd, sn, 63.5f);

  v8f acc = {};                                // 16x16 f32 accumulator
  v2f ones = {1.0f, 1.0f};                     // B = all-ones -> row-sum reduce

#pragma unroll 4
  for (int chunk = 0; chunk < 32; ++chunk) {
    const float tb = (float)((chunk << 2) + (hh << 1)) - 63.5f;
    v2f aop;
    aop[0] = bsample(img, uc, vc, cs, sn, tb);
    aop[1] = bsample(img, uc, vc, cs, sn, tb + 1.0f);
    acc = __builtin_amdgcn_wmma_f32_16x16x4_f32(
        /*neg_a=*/false, aop, /*neg_b=*/false, ones,
        /*c_mod=*/(short)0, acc, /*reuse_a=*/false, /*reuse_b=*/false);
  }

  // Column N=0 (lanes 0 and 16) holds the 16 per-angle ray sums.
  if (m == 0) {
    const int abase = base + (hh << 3);
    float* __restrict__ op = out + (size_t)(b * kC + c) * kA;
#pragma unroll
    for (int r = 0; r < 8; ++r) {
      const int a = abase + r;
      if (a < kA) op[a] = acc[r];
    }
  }
}

extern "C" void kernel_launch(void* const* d_in, const int* in_sizes, int n_in,
                              void* d_out, int out_size, void* d_ws, size_t ws_size,
                              hipStream_t stream) {
  const float* X      = (const float*)d_in[0];  // [8,128,128,128] f32
  const float* angles = (const float*)d_in[1];  // [180] f32
  float* out          = (float*)d_out;          // [8,128,1,180] f32

  dim3 grid(kC, kB);                            // one WG per (b, c) image
  radon_diag_kernel<<<grid, kThreads, 0, stream>>>(X, angles, out);
}